// GCN_36584531427972
// MI455X (gfx1250) — compile-verified
//
#include <hip/hip_runtime.h>
#include <hip/hip_bf16.h>
#include <math.h>

// ---------------------------------------------------------------------------
// Problem constants (from the reference file)
// ---------------------------------------------------------------------------
#define NN 100000      // nodes  (== 6250 * 16, so M-tiling is exact)
#define EE 3200000     // edges
#define FF 128         // input features
#define HH 64          // hidden features
#define GG 512         // graphs
#define OO 10          // outputs

typedef float v8f __attribute__((ext_vector_type(8)));
typedef float v2f __attribute__((ext_vector_type(2)));

// ---------------------------------------------------------------------------
// Degree / normalization
// ---------------------------------------------------------------------------
__global__ void k_init_deg(float* deg) {
    int i = blockIdx.x * blockDim.x + threadIdx.x;
    if (i < NN) deg[i] = 1.0f;              // self-loop contribution
}

__global__ void k_deg_scatter(const long long* dst, float* deg) {
    int e = blockIdx.x * blockDim.x + threadIdx.x;
    if (e < EE) atomicAdd(&deg[(int)dst[e]], 1.0f);
}

__global__ void k_deg_to_dis(float* deg) {
    int i = blockIdx.x * blockDim.x + threadIdx.x;
    if (i < NN) {
        float d = deg[i];
        deg[i] = (d > 0.0f) ? rsqrtf(d) : 0.0f;   // deg >= 1 always (self loops)
    }
}

// ---------------------------------------------------------------------------
// Dense GEMM via f32 WMMA:  Out[N x 64] = A[N x K] @ W[K x 64]
// One wave per 16x16 output tile; W cached in LDS; K stepped by 4 per WMMA.
// ---------------------------------------------------------------------------
template <int K>
__global__ __launch_bounds__(256) void k_gemm_wmma(const float* __restrict__ A,
                                                   const float* __restrict__ W,
                                                   float* __restrict__ Out) {
    __shared__ float ldsW[K * HH];          // <= 128*64*4 = 32 KB
    for (int i = threadIdx.x; i < K * HH; i += 256) ldsW[i] = W[i];
    __syncthreads();

    const int lane = threadIdx.x & 31;
    const int wv   = threadIdx.x >> 5;      // 8 waves / block
    const int tile = blockIdx.x * 8 + wv;   // tiles = (NN/16) * 4
    const int ntiles = (NN / 16) * 4;
    if (tile >= ntiles) return;

    const int tm = tile >> 2;               // M tile
    const int tn = tile & 3;                // N tile (4 tiles of 16 cover H=64)
    const int l15  = lane & 15;
    const bool hi  = lane >= 16;
    const int row  = tm * 16 + l15;         // A row for this lane
    const int col  = tn * 16 + l15;         // B/out column for this lane

    v8f acc = {};
    #pragma unroll
    for (int kk = 0; kk < K; kk += 4) {
        const int ka = kk + (hi ? 2 : 0);
        // A fragment (16x4): lane holds A[row][ka], A[row][ka+1]
        v2f af = { A[row * K + ka], A[row * K + ka + 1] };
        // B fragment (4x16): lane holds W[ka][col], W[ka+1][col]
        v2f bf = { ldsW[ka * HH + col], ldsW[(ka + 1) * HH + col] };
        acc = __builtin_amdgcn_wmma_f32_16x16x4_f32(
            /*neg_a=*/false, af, /*neg_b=*/false, bf,
            /*c_mod=*/(short)0, acc, /*reuse_a=*/false, /*reuse_b=*/false);
    }

    // C/D layout: VGPR v, lanes 0-15 -> M=v, lanes 16-31 -> M=v+8, N=lane&15
    const int mbase = tm * 16 + (hi ? 8 : 0);
    #pragma unroll
    for (int v = 0; v < 8; ++v)
        Out[(mbase + v) * HH + col] = acc[v];
}

// ---------------------------------------------------------------------------
// agg[i] = lin[i] * dis[i]^2   (self loop term, also zero-initializes agg)
// ---------------------------------------------------------------------------
__global__ void k_selfloop(const float* __restrict__ lin,
                           const float* __restrict__ dis,
                           float* __restrict__ agg) {
    long long idx = (long long)blockIdx.x * blockDim.x + threadIdx.x;
    if (idx < (long long)NN * HH) {
        int node = (int)(idx >> 6);
        float d = dis[node];
        agg[idx] = lin[idx] * d * d;
    }
}

// ---------------------------------------------------------------------------
// Edge scatter: one wave per edge, 2 features per lane, f32 atomics into L2
// ---------------------------------------------------------------------------
__global__ void k_edge_scatter(const long long* __restrict__ src,
                               const long long* __restrict__ dst,
                               const float* __restrict__ dis,
                               const float* __restrict__ lin,
                               float* __restrict__ agg) {
    long long gtid = (long long)blockIdx.x * blockDim.x + threadIdx.x;
    int e    = (int)(gtid >> 5);
    int lane = threadIdx.x & 31;
    if (e >= EE) return;
    int s = (int)src[e];
    int d = (int)dst[e];
    float w = dis[s] * dis[d];
    float m0 = lin[s * HH + lane]      * w;
    float m1 = lin[s * HH + lane + 32] * w;
    atomicAdd(&agg[d * HH + lane],      m0);
    atomicAdd(&agg[d * HH + lane + 32], m1);
}

// ---------------------------------------------------------------------------
// h = tanh(agg + bias)
// ---------------------------------------------------------------------------
__global__ void k_bias_tanh(const float* __restrict__ agg,
                            const float* __restrict__ bias,
                            float* __restrict__ h) {
    long long idx = (long long)blockIdx.x * blockDim.x + threadIdx.x;
    if (idx < (long long)NN * HH)
        h[idx] = tanhf(agg[idx] + bias[idx & (HH - 1)]);
}

// ---------------------------------------------------------------------------
// Pooling
// ---------------------------------------------------------------------------
__global__ void k_pool_init(float* maxp, float* sump, float* cnt) {
    int i = blockIdx.x * blockDim.x + threadIdx.x;
    if (i < GG * HH) {
        ((unsigned*)maxp)[i] = 0xFF800000u;   // -inf bit pattern
        sump[i] = 0.0f;
    }
    if (i < GG) cnt[i] = 0.0f;
}

__device__ __forceinline__ void atomicMaxF32(float* addr, float val) {
    if (val >= 0.0f)
        atomicMax((int*)addr, __float_as_int(val));
    else
        atomicMin((unsigned int*)addr, (unsigned int)__float_as_int(val));
}

__global__ void k_pool_scatter(const long long* __restrict__ batch,
                               const float* __restrict__ h,
                               float* __restrict__ maxp,
                               float* __restrict__ sump,
                               float* __restrict__ cnt) {
    long long gtid = (long long)blockIdx.x * blockDim.x + threadIdx.x;
    int node = (int)(gtid >> 5);
    int lane = threadIdx.x & 31;
    if (node >= NN) return;
    int g = (int)batch[node];
    float v0 = h[node * HH + lane];
    float v1 = h[node * HH + lane + 32];
    atomicAdd(&sump[g * HH + lane],      v0);
    atomicAdd(&sump[g * HH + lane + 32], v1);
    atomicMaxF32(&maxp[g * HH + lane],      v0);
    atomicMaxF32(&maxp[g * HH + lane + 32], v1);
    if (lane == 0) atomicAdd(&cnt[g], 1.0f);
}

// ---------------------------------------------------------------------------
// Final FC: out[g] = concat(maxp[g], sump[g]/cnt[g]) @ Wfc + bfc
// ---------------------------------------------------------------------------
__global__ void k_fc(const float* __restrict__ maxp,
                     const float* __restrict__ sump,
                     const float* __restrict__ cnt,
                     const float* __restrict__ Wfc,
                     const float* __restrict__ bfc,
                     float* __restrict__ out) {
    int g = blockIdx.x;
    int t = threadIdx.x;                        // 128 threads
    __shared__ float pooled[2 * HH];
    float v;
    if (t < HH) v = __int_as_float(((const int*)maxp)[g * HH + t]);
    else        v = sump[g * HH + (t - HH)] / fmaxf(cnt[g], 1.0f);
    pooled[t] = v;
    __syncthreads();
    if (t < OO) {
        float acc = bfc[t];
        #pragma unroll 8
        for (int j = 0; j < 2 * HH; ++j) acc += pooled[j] * Wfc[j * OO + t];
        out[g * OO + t] = acc;
    }
}

// ---------------------------------------------------------------------------
// Launch
// ---------------------------------------------------------------------------
extern "C" void kernel_launch(void* const* d_in, const int* in_sizes, int n_in,
                              void* d_out, int out_size, void* d_ws, size_t ws_size,
                              hipStream_t stream) {
    const float*     x     = (const float*)d_in[0];
    const long long* edge  = (const long long*)d_in[1];   // [2, E]
    const long long* batch = (const long long*)d_in[2];   // [N]
    const float* W1  = (const float*)d_in[4];
    const float* b1  = (const float*)d_in[5];
    const float* W2  = (const float*)d_in[6];
    const float* b2  = (const float*)d_in[7];
    const float* W3  = (const float*)d_in[8];
    const float* b3  = (const float*)d_in[9];
    const float* W4  = (const float*)d_in[10];
    const float* b4  = (const float*)d_in[11];
    const float* Wfc = (const float*)d_in[12];
    const float* bfc = (const float*)d_in[13];

    const long long* src = edge;
    const long long* dst = edge + EE;

    float* out  = (float*)d_out;            // [G, O]
    float* hout = (float*)d_out + GG * OO;  // [N, H]  (final h, tuple element 2)

    // Workspace layout (float offsets)
    float* ws   = (float*)d_ws;
    float* dis  = ws;                                  // N      (deg then rsqrt)
    float* lin  = ws + 102400;                         // N*H
    float* agg  = lin + (size_t)NN * HH;               // N*H
    float* hbuf = agg + (size_t)NN * HH;               // N*H
    float* maxp = hbuf + (size_t)NN * HH;              // G*H
    float* sump = maxp + GG * HH;                      // G*H
    float* cnt  = sump + GG * HH;                      // G

    const int T = 256;
    const int gN   = (NN + T - 1) / T;
    const int gE   = (EE + T - 1) / T;
    const int gNH  = (NN * HH + T - 1) / T;
    const int gEw  = (EE * 32 + T - 1) / T;            // wave per edge
    const int gNw  = (NN * 32 + T - 1) / T;            // wave per node
    const int gGem = ((NN / 16) * 4 + 7) / 8;          // 8 tiles (waves) / block

    // Normalization
    k_init_deg   <<<gN, T, 0, stream>>>(dis);
    k_deg_scatter<<<gE, T, 0, stream>>>(dst, dis);
    k_deg_to_dis <<<gN, T, 0, stream>>>(dis);

    // Layer 1 (K = 128)
    k_gemm_wmma<FF><<<gGem, T, 0, stream>>>(x, W1, lin);
    k_selfloop     <<<gNH, T, 0, stream>>>(lin, dis, agg);
    k_edge_scatter <<<gEw, T, 0, stream>>>(src, dst, dis, lin, agg);
    k_bias_tanh    <<<gNH, T, 0, stream>>>(agg, b1, hbuf);

    // Layer 2
    k_gemm_wmma<HH><<<gGem, T, 0, stream>>>(hbuf, W2, lin);
    k_selfloop     <<<gNH, T, 0, stream>>>(lin, dis, agg);
    k_edge_scatter <<<gEw, T, 0, stream>>>(src, dst, dis, lin, agg);
    k_bias_tanh    <<<gNH, T, 0, stream>>>(agg, b2, hbuf);

    // Layer 3
    k_gemm_wmma<HH><<<gGem, T, 0, stream>>>(hbuf, W3, lin);
    k_selfloop     <<<gNH, T, 0, stream>>>(lin, dis, agg);
    k_edge_scatter <<<gEw, T, 0, stream>>>(src, dst, dis, lin, agg);
    k_bias_tanh    <<<gNH, T, 0, stream>>>(agg, b3, hbuf);

    // Layer 4 -> writes h directly into d_out
    k_gemm_wmma<HH><<<gGem, T, 0, stream>>>(hbuf, W4, lin);
    k_selfloop     <<<gNH, T, 0, stream>>>(lin, dis, agg);
    k_edge_scatter <<<gEw, T, 0, stream>>>(src, dst, dis, lin, agg);
    k_bias_tanh    <<<gNH, T, 0, stream>>>(agg, b4, hout);

    // Pooling + FC
    k_pool_init   <<<(GG * HH + T - 1) / T, T, 0, stream>>>(maxp, sump, cnt);
    k_pool_scatter<<<gNw, T, 0, stream>>>(batch, hout, maxp, sump, cnt);
    k_fc          <<<GG, 128, 0, stream>>>(maxp, sump, cnt, Wfc, bfc, out);
}